// PPGNConv_78572131713434
// MI455X (gfx1250) — compile-verified
//
#include <hip/hip_runtime.h>

typedef __attribute__((ext_vector_type(2))) float v2f;
typedef __attribute__((ext_vector_type(4))) float v4f;
typedef __attribute__((ext_vector_type(8))) float v8f;

#define BN   128   // N
#define IND  32    // INDIM
#define OUTD 64    // OUTDIM
#define NB   32    // batch

// gfx1250 async global->LDS copy (per-lane LDS dest), tracked by ASYNCcnt
__device__ __forceinline__ void async_copy_b32(unsigned int lds_addr, const float* gptr) {
    asm volatile("global_load_async_to_lds_b32 %0, %1, off"
                 :: "v"(lds_addr), "v"((unsigned long long)(uintptr_t)gptr)
                 : "memory");
}
#define S_WAIT_ASYNCCNT_16() asm volatile("s_wait_asynccnt 0x10" ::: "memory")
#define S_WAIT_ASYNCCNT_0()  asm volatile("s_wait_asynccnt 0x0"  ::: "memory")

// ---------------------------------------------------------------------------
// Kernel 1: X1 = relu(X @ W1 + b1) * mask ; X2 = relu(X @ W2 + b2) * mask
// One block per 16 rows of flattened M = B*N*N. 8 waves: wave = which*4 + etile
// WMMA f32 16x16x4, K = 32 -> 8 steps.
// ---------------------------------------------------------------------------
__global__ __launch_bounds__(256) void ppgn_mlp(
    const float* __restrict__ X,
    const float* __restrict__ W1, const float* __restrict__ B1,
    const float* __restrict__ W2, const float* __restrict__ B2,
    const long long* __restrict__ n_nodes,
    float* __restrict__ X1, float* __restrict__ X2)
{
    __shared__ float xs[16 * 34];          // 16 X rows, padded stride 34
    __shared__ float wsm[2][OUTD * 34];    // W transposed: [which][e*34 + d]

    const int t = threadIdx.x;
    const long long m0 = (long long)blockIdx.x * 16;

    // stage 16 X rows (512 floats) as float2, coalesced
    {
        int f = t * 2;                 // 0..510
        int r = f >> 5, d = f & 31;
        v2f v = *(const v2f*)(X + (m0 + r) * IND + d);
        *(v2f*)(&xs[r * 34 + d]) = v;
    }
    // stage both weight matrices transposed to [e][d]
    for (int f = t; f < 2 * IND * OUTD; f += 256) {
        int which = f >> 11;
        int g = f & 2047;
        int d = g >> 6, e = g & 63;
        const float* Wsrc = which ? W2 : W1;
        wsm[which][e * 34 + d] = Wsrc[g];
    }
    __syncthreads();

    const int wave = t >> 5, lane = t & 31;
    const int which = wave >> 2;           // 0 -> W1/X1, 1 -> W2/X2
    const int e0 = (wave & 3) * 16;        // column tile
    const int half = lane >> 4, lrow = lane & 15;

    v8f acc = {};
    const float* wp = &wsm[which][0];
    #pragma unroll
    for (int s = 0; s < 8; ++s) {
        // A VGPR0: K=0 (lanes 0-15) / K=2 (lanes 16-31); VGPR1: K=1 / K=3
        int ks = 4 * s + 2 * half;
        v2f a = *(const v2f*)(&xs[lrow * 34 + ks]);
        v2f b = *(const v2f*)(&wp[(e0 + lrow) * 34 + ks]);
        acc = __builtin_amdgcn_wmma_f32_16x16x4_f32(
            false, a, false, b, (short)0, acc, false, false);
    }

    // epilogue: bias + relu + mask; rows m0..m0+15 share (b, i)
    const int bidx = (int)(m0 >> 14);          // / (N*N)
    const int irow = (int)((m0 >> 7) & 127);
    const int jb   = (int)(m0 & 127);
    const long long nn = n_nodes[bidx];
    const float im = (irow < nn) ? 1.0f : 0.0f;
    const float* bp = which ? B2 : B1;
    const float bias = bp[e0 + lrow];          // C col = lrow
    float* dst = which ? X2 : X1;
    #pragma unroll
    for (int v = 0; v < 8; ++v) {
        int r = v + 8 * half;                  // C row
        float val = acc[v] + bias;
        val = fmaxf(val, 0.0f);
        float jm = ((jb + r) < nn) ? 1.0f : 0.0f;
        val *= im * jm;
        dst[(m0 + r) * OUTD + e0 + lrow] = val;
    }
}

// ---------------------------------------------------------------------------
// Kernel 2: out[b,i,j,e] = sum_k X1[b,i,k,e] * X2[b,k,j,e]
// One block per (b, 16x16 i/j tile, e-half of 32 channels). 8 waves x 4 ch.
// Double-buffered LDS stages filled with GLOBAL_LOAD_ASYNC_TO_LDS_B32
// (per-lane LDS scatter into transposed [e][row][k] layout, stride 66 ->
// 8B-aligned v2f fragment loads, conflict-free). s_wait_asynccnt pipelines
// next-stage copies under the current stage's WMMAs.
// ---------------------------------------------------------------------------
#define KC    4
#define EH    32   // e-channels per block
#define EST   66

__global__ __launch_bounds__(256) void ppgn_fwl(
    const float* __restrict__ X1, const float* __restrict__ X2,
    float* __restrict__ Out)
{
    __shared__ float Als[2][EH * EST];   // [buf][e][i*4 + k]
    __shared__ float Bls[2][EH * EST];   // [buf][e][j*4 + k]

    const int t   = threadIdx.x;
    const int bid = blockIdx.x;
    const int eh  = bid & 1;
    const int jt  = (bid >> 1) & 7;
    const int it  = (bid >> 4) & 7;
    const int b   = bid >> 7;
    const int i0 = it * 16, j0 = jt * 16, e0 = eh * EH;

    const int wave = t >> 5, lane = t & 31;
    const int half = lane >> 4, lrow = lane & 15;

    // global bases include i0/j0 and the e-half
    const long long baseA = (((long long)b * BN + i0) * BN) * OUTD + e0; // +k*OUTD
    const long long baseB = (((long long)b * BN) * BN + j0) * OUTD + e0; // +k*BN*OUTD

    const unsigned int lbA0 = (unsigned int)(uintptr_t)(&Als[0][0]);
    const unsigned int lbA1 = (unsigned int)(uintptr_t)(&Als[1][0]);
    const unsigned int lbB0 = (unsigned int)(uintptr_t)(&Bls[0][0]);
    const unsigned int lbB1 = (unsigned int)(uintptr_t)(&Bls[1][0]);

    // 8 async B32 copies per thread per matrix per stage (16 per wave-stage)
    auto issue_stage = [&](int buf, int k0) {
        const unsigned int lA = buf ? lbA1 : lbA0;
        const unsigned int lB = buf ? lbB1 : lbB0;
        const float* gA = X1 + baseA + (long long)k0 * OUTD;
        const float* gB = X2 + baseB + (long long)k0 * (BN * OUTD);
        #pragma unroll
        for (int q = 0; q < 8; ++q) {
            int f = t + 256 * q;                       // 0..2047
            int e = f & 31, k = (f >> 5) & 3, i = f >> 7;
            async_copy_b32(lA + (unsigned int)(e * EST + i * KC + k) * 4u,
                           gA + (long long)i * (BN * OUTD) + k * OUTD + e);
        }
        #pragma unroll
        for (int q = 0; q < 8; ++q) {
            int f = t + 256 * q;                       // 0..2047
            int e = f & 31, j = (f >> 5) & 15, k = f >> 9;
            async_copy_b32(lB + (unsigned int)(e * EST + j * KC + k) * 4u,
                           gB + (long long)k * (BN * OUTD) + j * OUTD + e);
        }
    };

    v8f acc[4] = {};

    issue_stage(0, 0);
    const int NCHUNK = BN / KC;                        // 32
    for (int c = 0; c < NCHUNK; ++c) {
        const int cur = c & 1;
        if (c + 1 < NCHUNK) {
            issue_stage(cur ^ 1, (c + 1) * KC);        // prefetch next stage
            S_WAIT_ASYNCCNT_16();                      // current stage landed
        } else {
            S_WAIT_ASYNCCNT_0();
        }
        __syncthreads();                               // all waves' copies visible

        const int fo = lrow * KC + 2 * half;           // (K0,K1)/(K2,K3) halves
        #pragma unroll
        for (int ee = 0; ee < 4; ++ee) {
            int e = wave * 4 + ee;
            v2f a  = *(const v2f*)(&Als[cur][e * EST + fo]);
            v2f bb = *(const v2f*)(&Bls[cur][e * EST + fo]);
            acc[ee] = __builtin_amdgcn_wmma_f32_16x16x4_f32(
                false, a, false, bb, (short)0, acc[ee], false, false);
        }
        __syncthreads();                               // reads done before rewrite
    }

    // store: C row r = v + 8*half, col c = lrow; masked inputs => masked output
    const long long obase = (((long long)b * BN + i0) * BN + j0) * OUTD;
    #pragma unroll
    for (int ee = 0; ee < 4; ++ee) {
        int e = e0 + wave * 4 + ee;
        #pragma unroll
        for (int v = 0; v < 8; ++v) {
            int r = v + 8 * half;
            Out[obase + (long long)r * (BN * OUTD) + (long long)lrow * OUTD + e]
                = acc[ee][v];
        }
    }
}

// ---------------------------------------------------------------------------
extern "C" void kernel_launch(void* const* d_in, const int* in_sizes, int n_in,
                              void* d_out, int out_size, void* d_ws, size_t ws_size,
                              hipStream_t stream) {
    (void)in_sizes; (void)n_in; (void)out_size; (void)ws_size;
    const float*     X  = (const float*)d_in[0];
    const float*     W1 = (const float*)d_in[1];
    const float*     b1 = (const float*)d_in[2];
    const float*     W2 = (const float*)d_in[3];
    const float*     b2 = (const float*)d_in[4];
    const long long* nn = (const long long*)d_in[5];
    float* Out = (float*)d_out;

    float* X1 = (float*)d_ws;
    float* X2 = X1 + (size_t)NB * BN * BN * OUTD;

    dim3 blk(256);
    ppgn_mlp<<<(NB * BN * BN) / 16, blk, 0, stream>>>(X, W1, b1, W2, b2, nn, X1, X2);
    ppgn_fwl<<<NB * (BN / 16) * (BN / 16) * 2, blk, 0, stream>>>(X1, X2, Out);
}